// PointerGeneratorNetwork_17506286698897
// MI455X (gfx1250) — compile-verified
//
#include <hip/hip_runtime.h>
#include <hip/hip_bf16.h>
#include <math.h>

// ---------------- model dims ----------------
#define BB 16
#define SS 400
#define TT 100
#define VV 50000
#define MAX_OOV 50
#define EXT 50050
#define EE 128
#define HH 256
#define H2 512
#define H4 1024

typedef float v2f __attribute__((ext_vector_type(2)));
typedef float v8f __attribute__((ext_vector_type(8)));

__device__ inline v8f wmma4(v2f a, v2f b, v8f c) {
  // V_WMMA_F32_16X16X4_F32 : D(16x16 f32) = A(16x4) * B(4x16) + C
  return __builtin_amdgcn_wmma_f32_16x16x4_f32(
      false, a, false, b, (short)0, c, false, false);
}

__device__ inline float sigm(float x) { return 1.0f / (1.0f + __expf(-x)); }

// ---------------- generic WMMA f32 GEMM: C = A[MxK] @ W[KxN] ----------------
// Each wave computes NT consecutive 16x16 tiles of C, reusing the A fragment
// NT times per k-step (A loads amortized; B stream prefetched 32 rows ahead).
#define GF_BIAS 1
#define GF_ACC  2
#define GF_RELU 4

template <int NT>
__global__ void gemm_wmma_nt(const float* __restrict__ A, const float* __restrict__ W,
                             const float* __restrict__ bias, float* __restrict__ C,
                             int M, int K, int N, int flags) {
  int wave = (int)((blockIdx.x * blockDim.x + threadIdx.x) >> 5);
  int lane = threadIdx.x & 31;
  int tilesN = N >> 4;
  int groupsN = tilesN / NT;
  int tilesM = M >> 4;
  if (wave >= tilesM * groupsN) return;  // wave-uniform: surviving waves keep EXEC all-1s
  int tm = wave / groupsN, tg = wave % groupsN;
  int m0 = tm << 4;
  int n0 = (tg * NT) << 4;
  int half = lane >> 4;                  // K-half selector (A) / row-half (B)
  int l15 = lane & 15;

  const v8f vzero = {0.f, 0.f, 0.f, 0.f, 0.f, 0.f, 0.f, 0.f};
  v8f acc[NT];
#pragma unroll
  for (int nt = 0; nt < NT; ++nt) {
    if (flags & GF_ACC) {
#pragma unroll
      for (int r = 0; r < 8; ++r)
        acc[nt][r] = C[(size_t)(m0 + r + half * 8) * N + n0 + nt * 16 + l15];
    } else {
      acc[nt] = vzero;
    }
  }

  const float* Ar = A + (size_t)(m0 + l15) * K;
  const float* Wc = W + n0 + l15;
  for (int k = 0; k < K; k += 4) {
    v2f a;
    a.x = Ar[k + 2 * half];
    a.y = Ar[k + 2 * half + 1];
    const float* w0 = Wc + (size_t)(k + 2 * half) * N;
    const float* w1 = w0 + N;
    if (k + 32 < K)                       // stays in-bounds: row <= K-2
      __builtin_prefetch(w0 + (size_t)32 * N, 0, 3);   // global_prefetch_b8
#pragma unroll
    for (int nt = 0; nt < NT; ++nt) {
      v2f b;
      b.x = w0[nt * 16];
      b.y = w1[nt * 16];
      acc[nt] = wmma4(a, b, acc[nt]);
    }
  }

#pragma unroll
  for (int nt = 0; nt < NT; ++nt) {
    float bv = (flags & GF_BIAS) ? bias[n0 + nt * 16 + l15] : 0.f;
#pragma unroll
    for (int r = 0; r < 8; ++r) {
      float v = acc[nt][r] + bv;
      if (flags & GF_RELU) v = fmaxf(v, 0.f);
      C[(size_t)(m0 + r + half * 8) * N + n0 + nt * 16 + l15] = v;
    }
  }
}

// ---------------- encoder embedding gather (time-major S,B,E) ----------------
__global__ void embed_enc(const float* __restrict__ emb, const int* __restrict__ ids,
                          float* __restrict__ out) {
  int i = blockIdx.x * blockDim.x + threadIdx.x;   // over S*B*E
  if (i >= SS * BB * EE) return;
  int e = i % EE;
  int sb = i / EE;
  int b = sb % BB;
  int s = sb / BB;
  out[i] = emb[(size_t)ids[b * SS + s] * EE + e];
}

// ---------------- persistent LSTM scan (one workgroup per direction) ----------------
// 512 threads = 16 waves; wave w owns hidden units [16w,16w+16); c in WMMA regs; h in LDS.
#define HP 260   // padded LDS row stride (floats) to avoid 16-way bank conflicts

__global__ void __launch_bounds__(512)
lstm_scan(const float* __restrict__ xWx_f, const float* __restrict__ Wh_f,
          const float* __restrict__ xWx_b, const float* __restrict__ Wh_b,
          float* __restrict__ hs_f, float* __restrict__ hs_b,
          float* __restrict__ hT, float* __restrict__ cT) {
  __shared__ float hsh[BB * HP];
  const int dir = blockIdx.x;
  const float* xWx = dir ? xWx_b : xWx_f;
  const float* Wh  = dir ? Wh_b  : Wh_f;
  float* hs        = dir ? hs_b  : hs_f;

  int tid = threadIdx.x;
  int wave = tid >> 5, lane = tid & 31;
  int half = lane >> 4, l15 = lane & 15;
  int n0 = wave << 4;                       // hidden-unit tile base

  for (int i = tid; i < BB * HP; i += blockDim.x) hsh[i] = 0.f;
  v8f c = {0.f, 0.f, 0.f, 0.f, 0.f, 0.f, 0.f, 0.f};
  __syncthreads();

  for (int t = 0; t < SS; ++t) {
    int idx = dir ? (SS - 1 - t) : t;       // bwd reads reversed, writes to final order
    const float* xrow = xWx + (size_t)idx * BB * H4;

    v8f zi = {0.f,0.f,0.f,0.f,0.f,0.f,0.f,0.f};
    v8f zf = zi, zg = zi, zo = zi;
    for (int k = 0; k < HH; k += 4) {
      v2f a, b;
      a.x = hsh[l15 * HP + k + 2 * half];
      a.y = hsh[l15 * HP + k + 2 * half + 1];
      const float* Wr0 = Wh + (size_t)(k + 2 * half) * H4;
      const float* Wr1 = Wr0 + H4;
      b.x = Wr0[0 * HH + n0 + l15]; b.y = Wr1[0 * HH + n0 + l15]; zi = wmma4(a, b, zi);
      b.x = Wr0[1 * HH + n0 + l15]; b.y = Wr1[1 * HH + n0 + l15]; zf = wmma4(a, b, zf);
      b.x = Wr0[2 * HH + n0 + l15]; b.y = Wr1[2 * HH + n0 + l15]; zg = wmma4(a, b, zg);
      b.x = Wr0[3 * HH + n0 + l15]; b.y = Wr1[3 * HH + n0 + l15]; zo = wmma4(a, b, zo);
    }
    __syncthreads();   // all waves done reading h before it is rewritten
#pragma unroll
    for (int r = 0; r < 8; ++r) {
      int row = r + half * 8;               // batch index
      int col = n0 + l15;                   // hidden unit
      const float* xr = xrow + (size_t)row * H4;
      float vi = zi[r] + xr[0 * HH + col];
      float vf = zf[r] + xr[1 * HH + col];
      float vg = zg[r] + xr[2 * HH + col];
      float vo = zo[r] + xr[3 * HH + col];
      float cn = sigm(vf) * c[r] + sigm(vi) * tanhf(vg);
      float hn = sigm(vo) * tanhf(cn);
      c[r] = cn;
      hsh[row * HP + col] = hn;
      hs[((size_t)idx * BB + row) * HH + col] = hn;
    }
    __syncthreads();
  }
  // final states -> concat buffers (B x 2H), dir selects half
#pragma unroll
  for (int r = 0; r < 8; ++r) {
    int row = r + half * 8, col = n0 + l15;
    hT[row * H2 + dir * HH + col] = hsh[row * HP + col];
    cT[row * H2 + dir * HH + col] = c[r];
  }
}

// ---------------- enc_out assembly: (B,S,2H) from time-major h streams ----------------
__global__ void assemble_enc_out(const float* __restrict__ hs_f, const float* __restrict__ hs_b,
                                 float* __restrict__ enc_out) {
  int i = blockIdx.x * blockDim.x + threadIdx.x;   // B*S*2H
  if (i >= BB * SS * H2) return;
  int j = i % H2;
  int bs = i / H2;
  int s = bs % SS;
  int b = bs / SS;
  enc_out[i] = (j < HH) ? hs_f[((size_t)s * BB + b) * HH + j]
                        : hs_b[((size_t)s * BB + b) * HH + (j - HH)];
}

// ---------------- misc small kernels ----------------
__global__ void zero_kernel(float* p, int n) {
  int i = blockIdx.x * blockDim.x + threadIdx.x;
  if (i < n) p[i] = 0.f;
}

__global__ void mask_sum_kernel(const float* __restrict__ mask, float* msum) {
  __shared__ float red[256];
  int i = blockIdx.x * blockDim.x + threadIdx.x;
  float v = (i < BB * SS) ? mask[i] : 0.f;
  red[threadIdx.x] = v; __syncthreads();
  for (int st = 128; st; st >>= 1) { if (threadIdx.x < st) red[threadIdx.x] += red[threadIdx.x + st]; __syncthreads(); }
  if (threadIdx.x == 0) atomicAdd(msum, red[0]);
}

__global__ void dec_prep(const float* __restrict__ emb, const int* __restrict__ dec_in,
                         const float* __restrict__ ctx, float* __restrict__ xc, int t) {
  int i = blockIdx.x * blockDim.x + threadIdx.x;   // 16*640
  if (i >= BB * (EE + H2)) return;
  int b = i / (EE + H2), j = i % (EE + H2);
  xc[i] = (j < EE) ? emb[(size_t)dec_in[b * TT + t] * EE + j] : ctx[b * H2 + (j - EE)];
}

__global__ void dec_gates(const float* __restrict__ zg, float* __restrict__ h,
                          float* __restrict__ c, float* __restrict__ svec) {
  int i = blockIdx.x * blockDim.x + threadIdx.x;   // 16*256
  if (i >= BB * HH) return;
  int b = i / HH, j = i % HH;
  const float* z = zg + (size_t)b * H4;
  float vi = z[j], vf = z[HH + j], vg = z[2 * HH + j], vo = z[3 * HH + j];
  float cn = sigm(vf) * c[i] + sigm(vi) * tanhf(vg);
  float hn = sigm(vo) * tanhf(cn);
  c[i] = cn; h[i] = hn;
  svec[b * H2 + j] = hn;
  svec[b * H2 + HH + j] = cn;
}

__global__ void attn_e_kernel(const float* __restrict__ enc_feats, const float* __restrict__ sWs,
                              const float* __restrict__ cov, const float* __restrict__ wc,
                              const float* __restrict__ vvec, float* __restrict__ e) {
  int gw = (int)((blockIdx.x * blockDim.x + threadIdx.x) >> 5);
  int lane = threadIdx.x & 31;
  if (gw >= BB * SS) return;
  int b = gw / SS, s = gw % SS;
  const float* ef = enc_feats + ((size_t)b * SS + s) * H2;
  const float* sw = sWs + (size_t)b * H2;
  float cv = cov[b * SS + s];
  float acc = 0.f;
  for (int j = lane; j < H2; j += 32)
    acc += tanhf(ef[j] + sw[j] + cv * wc[j]) * vvec[j];
  for (int off = 16; off; off >>= 1) acc += __shfl_xor(acc, off);
  if (lane == 0) e[b * SS + s] = acc;
}

__global__ void attn_softmax(const float* __restrict__ e, const float* __restrict__ mask,
                             float* __restrict__ attn, float* __restrict__ cov,
                             float* covloss_acc) {
  __shared__ float red[256];
  int b = blockIdx.x, tid = threadIdx.x;
  const float* eb = e + (size_t)b * SS;
  const float* mb = mask + (size_t)b * SS;
  float m = -1e30f;
  for (int s = tid; s < SS; s += 256) m = fmaxf(m, eb[s]);
  red[tid] = m; __syncthreads();
  for (int st = 128; st; st >>= 1) { if (tid < st) red[tid] = fmaxf(red[tid], red[tid + st]); __syncthreads(); }
  m = red[0]; __syncthreads();
  float sum = 0.f;
  for (int s = tid; s < SS; s += 256) sum += __expf(eb[s] - m);
  red[tid] = sum; __syncthreads();
  for (int st = 128; st; st >>= 1) { if (tid < st) red[tid] += red[tid + st]; __syncthreads(); }
  float inv = 1.f / red[0]; __syncthreads();
  float msum = 0.f;
  for (int s = tid; s < SS; s += 256) msum += __expf(eb[s] - m) * inv * mb[s];
  red[tid] = msum; __syncthreads();
  for (int st = 128; st; st >>= 1) { if (tid < st) red[tid] += red[tid + st]; __syncthreads(); }
  float denom = red[0] + 1e-12f; __syncthreads();
  float closs = 0.f;
  for (int s = tid; s < SS; s += 256) {
    float a = __expf(eb[s] - m) * inv * mb[s] / denom;
    float cv = cov[b * SS + s];
    closs += fminf(a, cv) * mb[s];          // min(attn_t, cov_prev_t) * mask
    attn[b * SS + s] = a;
    cov[b * SS + s] = cv + a;
  }
  red[tid] = closs; __syncthreads();
  for (int st = 128; st; st >>= 1) { if (tid < st) red[tid] += red[tid + st]; __syncthreads(); }
  if (tid == 0) atomicAdd(covloss_acc, red[0]);
}

__global__ void attn_context(const float* __restrict__ attn, const float* __restrict__ enc_out,
                             float* __restrict__ ctx) {
  int b = blockIdx.x, j = threadIdx.x;     // 16 blocks x 512 threads
  const float* eo = enc_out + (size_t)b * SS * H2 + j;
  const float* at = attn + (size_t)b * SS;
  float acc = 0.f;
  for (int s = 0; s < SS; ++s) acc += at[s] * eo[(size_t)s * H2];
  ctx[b * H2 + j] = acc;
}

__global__ void hc_cat(const float* __restrict__ h, const float* __restrict__ ctx,
                       float* __restrict__ hc) {
  int i = blockIdx.x * blockDim.x + threadIdx.x;   // 16*768
  if (i >= BB * (HH + H2)) return;
  int b = i / (HH + H2), j = i % (HH + H2);
  hc[i] = (j < HH) ? h[b * HH + j] : ctx[b * H2 + (j - HH)];
}

__global__ void __launch_bounds__(1024) vocab_softmax(float* __restrict__ logits) {
  __shared__ float red[1024];
  int b = blockIdx.x, tid = threadIdx.x;
  float* row = logits + (size_t)b * VV;
  float m = -1e30f;
  for (int v = tid; v < VV; v += 1024) m = fmaxf(m, row[v]);
  red[tid] = m; __syncthreads();
  for (int st = 512; st; st >>= 1) { if (tid < st) red[tid] = fmaxf(red[tid], red[tid + st]); __syncthreads(); }
  m = red[0]; __syncthreads();
  float sum = 0.f;
  for (int v = tid; v < VV; v += 1024) sum += __expf(row[v] - m);
  red[tid] = sum; __syncthreads();
  for (int st = 512; st; st >>= 1) { if (tid < st) red[tid] += red[tid + st]; __syncthreads(); }
  float inv = 1.f / red[0]; __syncthreads();
  for (int v = tid; v < VV; v += 1024) row[v] = __expf(row[v] - m) * inv;
}

__global__ void pgen_kernel(const float* __restrict__ ctx, const float* __restrict__ svec,
                            const float* __restrict__ x_in, const float* __restrict__ W,
                            const float* __restrict__ bias, float* __restrict__ pgen) {
  int wave = threadIdx.x >> 5, lane = threadIdx.x & 31;   // 1 block x 512 thr, wave per batch
  int b = wave;
  float acc = 0.f;
  for (int j = lane; j < H2; j += 32) acc += ctx[b * H2 + j] * W[j];
  for (int j = lane; j < H2; j += 32) acc += svec[b * H2 + j] * W[H2 + j];
  for (int j = lane; j < EE; j += 32) acc += x_in[b * EE + j] * W[2 * H2 + j];
  for (int off = 16; off; off >>= 1) acc += __shfl_xor(acc, off);
  if (lane == 0) pgen[b] = sigm(acc + bias[0]);
}

__global__ void final_write(const float* __restrict__ pv, const float* __restrict__ pgen,
                            float* __restrict__ outF, int t) {
  size_t i = (size_t)blockIdx.x * blockDim.x + threadIdx.x;   // 16*50050
  if (i >= (size_t)BB * EXT) return;
  int b = (int)(i / EXT), v = (int)(i % EXT);
  float val = (v < VV) ? pgen[b] * pv[(size_t)b * VV + v] : 0.f;
  outF[((size_t)b * TT + t) * EXT + v] = val;   // (B,T,EXT) layout
}

__global__ void final_scatter(const float* __restrict__ attn, const float* __restrict__ pgen,
                              const int* __restrict__ ext_ids, float* __restrict__ outF, int t) {
  int i = blockIdx.x * blockDim.x + threadIdx.x;   // 16*400
  if (i >= BB * SS) return;
  int b = i / SS, s = i % SS;
  int id = ext_ids[b * SS + s];
  atomicAdd(&outF[((size_t)b * TT + t) * EXT + id], (1.f - pgen[b]) * attn[i]);
}

__global__ void loss_step(const float* __restrict__ outF, const int* __restrict__ tgt,
                          float* vnum, float* vcnt, int t) {
  int b = threadIdx.x;
  if (b >= BB) return;
  int tg = tgt[b * TT + t];
  int tc = min(max(tg, 0), EXT - 1);
  float p = outF[((size_t)b * TT + t) * EXT + tc] + 1e-12f;
  if (tg != 0) { atomicAdd(vnum, -logf(p)); atomicAdd(vcnt, 1.f); }
}

__global__ void finalize(const float* vnum, const float* vcnt, const float* covsum,
                         const float* msum, float* out3) {
  if (threadIdx.x == 0) {
    float vl = vnum[0] / vcnt[0];
    float cl = covsum[0] / msum[0];
    out3[0] = vl + cl;
    out3[1] = vl;
    out3[2] = cl;
  }
}

// ---------------- host orchestration ----------------
extern "C" void kernel_launch(void* const* d_in, const int* in_sizes, int n_in,
                              void* d_out, int out_size, void* d_ws, size_t ws_size,
                              hipStream_t stream) {
  (void)in_sizes; (void)n_in; (void)out_size; (void)ws_size;
  // params (dict insertion order), then other inputs
  const float* P_emb      = (const float*)d_in[0];
  const float* P_efWx     = (const float*)d_in[1];
  const float* P_efWh     = (const float*)d_in[2];
  const float* P_efb      = (const float*)d_in[3];
  const float* P_ebWx     = (const float*)d_in[4];
  const float* P_ebWh     = (const float*)d_in[5];
  const float* P_ebb      = (const float*)d_in[6];
  const float* P_redhW    = (const float*)d_in[7];
  const float* P_redhb    = (const float*)d_in[8];
  const float* P_redcW    = (const float*)d_in[9];
  const float* P_redcb    = (const float*)d_in[10];
  const float* P_xctxW    = (const float*)d_in[11];
  const float* P_xctxb    = (const float*)d_in[12];
  const float* P_decWx    = (const float*)d_in[13];
  const float* P_decWh    = (const float*)d_in[14];
  const float* P_decb     = (const float*)d_in[15];
  const float* P_attnWh   = (const float*)d_in[16];
  const float* P_attnWs   = (const float*)d_in[17];
  const float* P_attnwc   = (const float*)d_in[18];
  const float* P_attnb    = (const float*)d_in[19];
  const float* P_attnv    = (const float*)d_in[20];
  const float* P_outW1    = (const float*)d_in[21];
  const float* P_outb1    = (const float*)d_in[22];
  const float* P_outW2    = (const float*)d_in[23];
  const float* P_outb2    = (const float*)d_in[24];
  const float* P_pgenW    = (const float*)d_in[25];
  const float* P_pgenb    = (const float*)d_in[26];
  const int*   I_encIn    = (const int*)d_in[27];
  const int*   I_extIn    = (const int*)d_in[28];
  /* d_in[29] = encoder_length (unused, matches reference) */
  const float* I_encMask  = (const float*)d_in[30];
  const int*   I_decIn    = (const int*)d_in[31];
  const int*   I_decTgt   = (const int*)d_in[32];

  float* outAll = (float*)d_out;          // [loss, vloss, closs, finals(B,T,EXT)]
  float* outF = outAll + 3;

  // workspace carve-up
  float* ws = (float*)d_ws;
  size_t off = 0;
  auto alloc = [&](size_t n) { float* p = ws + off; off += n; return p; };
  float* emb_t   = alloc((size_t)SS * BB * EE);
  float* xWx_f   = alloc((size_t)SS * BB * H4);
  float* xWx_b   = alloc((size_t)SS * BB * H4);
  float* hs_f    = alloc((size_t)SS * BB * HH);
  float* hs_b    = alloc((size_t)SS * BB * HH);
  float* hTc     = alloc((size_t)BB * H2);
  float* cTc     = alloc((size_t)BB * H2);
  float* h_state = alloc((size_t)BB * HH);
  float* c_state = alloc((size_t)BB * HH);
  float* enc_out = alloc((size_t)BB * SS * H2);
  float* enc_ft  = alloc((size_t)BB * SS * H2);
  float* xc      = alloc((size_t)BB * (EE + H2));
  float* x_in    = alloc((size_t)BB * EE);
  float* zgbuf   = alloc((size_t)BB * H4);
  float* svec    = alloc((size_t)BB * H2);
  float* sWsbuf  = alloc((size_t)BB * H2);
  float* ebuf    = alloc((size_t)BB * SS);
  float* attnbuf = alloc((size_t)BB * SS);
  float* cov     = alloc((size_t)BB * SS);
  float* ctx     = alloc((size_t)BB * H2);
  float* hcbuf   = alloc((size_t)BB * (HH + H2));
  float* o1buf   = alloc((size_t)BB * HH);
  float* logits  = alloc((size_t)BB * VV);
  float* pgenbuf = alloc((size_t)BB);
  float* vnum    = alloc(1);
  float* vcnt    = alloc(1);
  float* covsum  = alloc(1);
  float* msum    = alloc(1);

  auto gemm = [&](const float* A, const float* W, const float* bias, float* C,
                  int M, int K, int N, int flags) {
    int tilesN = N / 16;
    if (tilesN % 4 == 0) {
      int waves = (M / 16) * (tilesN / 4);
      gemm_wmma_nt<4><<<(waves * 32 + 255) / 256, 256, 0, stream>>>(A, W, bias, C, M, K, N, flags);
    } else {                               // N = 50000 -> 3125 tiles = 5^5
      int waves = (M / 16) * (tilesN / 5);
      gemm_wmma_nt<5><<<(waves * 32 + 255) / 256, 256, 0, stream>>>(A, W, bias, C, M, K, N, flags);
    }
  };
  auto zero = [&](float* p, int n) {
    zero_kernel<<<(n + 255) / 256, 256, 0, stream>>>(p, n);
  };

  // ---- init running state / accumulators ----
  zero(cov, BB * SS);
  zero(ctx, BB * H2);
  zero(vnum, 4);                 // vnum, vcnt, covsum, msum are contiguous
  mask_sum_kernel<<<(BB * SS + 255) / 256, 256, 0, stream>>>(I_encMask, msum);

  // ---- encoder ----
  embed_enc<<<(SS * BB * EE + 255) / 256, 256, 0, stream>>>(P_emb, I_encIn, emb_t);
  gemm(emb_t, P_efWx, P_efb, xWx_f, SS * BB, EE, H4, GF_BIAS);
  gemm(emb_t, P_ebWx, P_ebb, xWx_b, SS * BB, EE, H4, GF_BIAS);
  lstm_scan<<<2, 512, 0, stream>>>(xWx_f, P_efWh, xWx_b, P_ebWh, hs_f, hs_b, hTc, cTc);
  gemm(hTc, P_redhW, P_redhb, h_state, BB, H2, HH, GF_BIAS | GF_RELU);
  gemm(cTc, P_redcW, P_redcb, c_state, BB, H2, HH, GF_BIAS | GF_RELU);
  assemble_enc_out<<<(BB * SS * H2 + 255) / 256, 256, 0, stream>>>(hs_f, hs_b, enc_out);
  gemm(enc_out, P_attnWh, nullptr, enc_ft, BB * SS, H2, H2, 0);

  // ---- decoder scan ----
  for (int t = 0; t < TT; ++t) {
    dec_prep<<<(BB * (EE + H2) + 255) / 256, 256, 0, stream>>>(P_emb, I_decIn, ctx, xc, t);
    gemm(xc, P_xctxW, P_xctxb, x_in, BB, EE + H2, EE, GF_BIAS);
    gemm(x_in, P_decWx, P_decb, zgbuf, BB, EE, H4, GF_BIAS);
    gemm(h_state, P_decWh, nullptr, zgbuf, BB, HH, H4, GF_ACC);
    dec_gates<<<(BB * HH + 255) / 256, 256, 0, stream>>>(zgbuf, h_state, c_state, svec);
    gemm(svec, P_attnWs, P_attnb, sWsbuf, BB, H2, H2, GF_BIAS);
    attn_e_kernel<<<(BB * SS * 32 + 255) / 256, 256, 0, stream>>>(
        enc_ft, sWsbuf, cov, P_attnwc, P_attnv, ebuf);
    attn_softmax<<<BB, 256, 0, stream>>>(ebuf, I_encMask, attnbuf, cov, covsum);
    attn_context<<<BB, H2, 0, stream>>>(attnbuf, enc_out, ctx);
    hc_cat<<<(BB * (HH + H2) + 255) / 256, 256, 0, stream>>>(h_state, ctx, hcbuf);
    gemm(hcbuf, P_outW1, P_outb1, o1buf, BB, HH + H2, HH, GF_BIAS);
    gemm(o1buf, P_outW2, P_outb2, logits, BB, HH, VV, GF_BIAS);
    vocab_softmax<<<BB, 1024, 0, stream>>>(logits);
    pgen_kernel<<<1, 512, 0, stream>>>(ctx, svec, x_in, P_pgenW, P_pgenb, pgenbuf);
    final_write<<<(BB * EXT + 255) / 256, 256, 0, stream>>>(logits, pgenbuf, outF, t);
    final_scatter<<<(BB * SS + 255) / 256, 256, 0, stream>>>(attnbuf, pgenbuf, I_extIn, outF, t);
    loss_step<<<1, 32, 0, stream>>>(outF, I_decTgt, vnum, vcnt, t);
  }
  finalize<<<1, 32, 0, stream>>>(vnum, vcnt, covsum, msum, outAll);
}